// DetectionLayer_52999896432949
// MI455X (gfx1250) — compile-verified
//
#include <hip/hip_runtime.h>
#include <cstdint>

// ---- problem constants (match reference) ----
#define BSZ    32
#define NANCH  90000            // FH*FW*NA = 100*100*9
#define KPRE   1000
#define KPAD   1024             // 64 tiles of 16 -> 32 full mask words
#define NT     64
#define KPOST  300
#define NMSTHR 0.7f
#define CAND   2048

typedef float v2f __attribute__((ext_vector_type(2)));
typedef float v8f __attribute__((ext_vector_type(8)));
typedef int   v4i __attribute__((ext_vector_type(4)));

#if defined(__has_builtin)
#  if __has_builtin(__builtin_amdgcn_global_load_async_to_lds_b128)
#    define HAVE_ASYNC_LDS 1
#  endif
#  if __has_builtin(__builtin_amdgcn_s_wait_asynccnt)
#    define HAVE_WAIT_ASYNC 1
#  endif
#endif

#ifdef HAVE_ASYNC_LDS
// b128 async copy payload pointers: AS1 (global, prints as __device__) source,
// AS3 (LDS, prints as __shared__) destination, per the builtin's prototype.
typedef __attribute__((address_space(1))) v4i* gv4i_p;
typedef __attribute__((address_space(3))) v4i* lv4i_p;
#endif

// ---------------- kernel 1: sigmoid scores (pure streaming, HBM-bound) ---------------
__global__ void score_sigmoid_kernel(const float4* __restrict__ logits,
                                     float4* __restrict__ scores, int n4) {
  int i = blockIdx.x * blockDim.x + threadIdx.x;
  if (i < n4) {
    float4 x = logits[i];
    float4 r;
    r.x = 1.0f / (1.0f + __expf(-x.x));
    r.y = 1.0f / (1.0f + __expf(-x.y));
    r.z = 1.0f / (1.0f + __expf(-x.z));
    r.w = 1.0f / (1.0f + __expf(-x.w));
    scores[i] = r;
  }
}

// ------- kernel 2: exact per-batch top-1000 via 2-level radix select + bitonic -------
// One workgroup (1024 threads = 32 wave32) per batch image.
// Scores are sigmoid outputs (positive floats) -> IEEE bit pattern is monotone.
__global__ __launch_bounds__(1024)
void topk_select_kernel(const float* __restrict__ scores,
                        unsigned* __restrict__ sel_idx,
                        float* __restrict__ sel_score) {
  __shared__ unsigned s_hist[2048];
  __shared__ unsigned long long s_cand[CAND];
  __shared__ unsigned s_b1, s_above1, s_thr22, s_cnt;
  const int b = blockIdx.x;
  const int tid = threadIdx.x;
  const float* sc = scores + (size_t)b * NANCH;

  for (int i = tid; i < 2048; i += 1024) s_hist[i] = 0;
  if (tid == 0) s_cnt = 0;
  __syncthreads();

  // level-1 histogram on top 11 bits
  for (int i = tid; i < NANCH; i += 1024) {
    unsigned u = __float_as_uint(sc[i]);
    atomicAdd(&s_hist[u >> 21], 1u);
  }
  __syncthreads();
  if (tid == 0) {
    unsigned acc = 0; int bin = 2047;
    for (; bin > 0; --bin) {
      if (acc + s_hist[bin] >= KPRE) break;
      acc += s_hist[bin];
    }
    s_b1 = (unsigned)bin; s_above1 = acc;
  }
  __syncthreads();
  const unsigned b1 = s_b1;

  // level-2 histogram on next 11 bits, within threshold bin
  for (int i = tid; i < 2048; i += 1024) s_hist[i] = 0;
  __syncthreads();
  for (int i = tid; i < NANCH; i += 1024) {
    unsigned u = __float_as_uint(sc[i]);
    if ((u >> 21) == b1) atomicAdd(&s_hist[(u >> 10) & 0x7FFu], 1u);
  }
  __syncthreads();
  if (tid == 0) {
    unsigned acc = s_above1; int bin = 2047;
    for (; bin > 0; --bin) {
      if (acc + s_hist[bin] >= KPRE) break;
      acc += s_hist[bin];
    }
    s_thr22 = (b1 << 11) | (unsigned)bin;
  }
  __syncthreads();
  const unsigned thr22 = s_thr22;

  // compact candidates (>= 1000, << 2048 for continuous scores)
  for (int i = tid; i < NANCH; i += 1024) {
    unsigned u = __float_as_uint(sc[i]);
    if ((u >> 10) >= thr22) {
      unsigned p = atomicAdd(&s_cnt, 1u);
      if (p < CAND)
        s_cand[p] = ((unsigned long long)u << 32) | (unsigned)(~(unsigned)i);
    }
  }
  __syncthreads();
  unsigned cnt = s_cnt; if (cnt > CAND) cnt = CAND;
  for (int i = tid; i < CAND; i += 1024)
    if (i >= (int)cnt) s_cand[i] = 0ull;
  __syncthreads();

  // bitonic sort, descending by (score_bits, ~index) -> score desc, index asc ties
  for (unsigned k = 2; k <= CAND; k <<= 1) {
    for (unsigned j = k >> 1; j > 0; j >>= 1) {
      for (unsigned i = tid; i < CAND; i += 1024) {
        unsigned p = i ^ j;
        if (p > i) {
          unsigned long long a = s_cand[i], c = s_cand[p];
          bool up = ((i & k) == 0);
          if ((a < c) == up) { s_cand[i] = c; s_cand[p] = a; }
        }
      }
      __syncthreads();
    }
  }
  if (tid < KPRE) {
    unsigned long long key = s_cand[tid];
    sel_score[b * KPRE + tid] = __uint_as_float((unsigned)(key >> 32));
    sel_idx[b * KPRE + tid]   = ~(unsigned)(key & 0xFFFFFFFFull);
  }
}

// ---- kernel 3: async-gather + decode survivors, WMMA-tiled IoU mask, greedy NMS ----
// One workgroup per batch. union(i,j) = area_i + area_j - inter(i,j) is computed as
// D = A x B + C with V_WMMA_F32_16X16X4_F32:
//   A (16x4): K0 row = area_i, K1 row = 1     (lanes 0-15 hold K0/K1, lanes16-31 K2/K3=0)
//   B (4x16): K0 row = 1,      K1 row = area_j
//   C        = -inter tile (C/D layout: VGPR r, lanes0-15 -> M=r, lanes16-31 -> M=r+8)
// IoU > thr  <=>  inter > thr * union   (division-free; union==0 -> false, matches ref)
// Each wave owns a full 32-bit mask word column (two adjacent 16-col tiles) per
// 16-row block -> ballot-packed plain stores, zero atomics, single writer per word.
__global__ __launch_bounds__(1024)
void nms_kernel(const float* __restrict__ deltas,
                const float* __restrict__ anchors,
                const unsigned* __restrict__ sel_idx,
                const float* __restrict__ sel_score,
                unsigned* __restrict__ mask_g,
                float* __restrict__ out,
                const int* __restrict__ p_img_h,
                const int* __restrict__ p_img_w) {
  __shared__ alignas(16) float s_box[KPAD][4];
  __shared__ alignas(16) float s_anc[KPAD][4];
  __shared__ float s_area[KPAD];
  __shared__ float s_sc[KPAD];
  __shared__ unsigned s_keep[32];
  const int b = blockIdx.x;
  const int tid = threadIdx.x;
  const float fw = (float)(*p_img_w);
  const float fh = (float)(*p_img_h);
  unsigned* mask = mask_g + (size_t)b * (KPAD * 32);

  // ---- stage the 1000 survivor deltas+anchors rows into LDS -----------------------
#ifdef HAVE_ASYNC_LDS
  // CDNA5 async gather: per-lane global addr -> per-lane LDS addr, ASYNCcnt-tracked.
  for (int i = tid; i < KPAD; i += 1024) {
    if (i < KPRE) {
      unsigned idx = sel_idx[b * KPRE + i];
      const float* dsrc = deltas + ((size_t)b * NANCH + idx) * 4;
      const float* asrc = anchors + (size_t)idx * 4;
      __builtin_amdgcn_global_load_async_to_lds_b128(
          (gv4i_p)dsrc, (lv4i_p)&s_box[i][0], 0, 0);
      __builtin_amdgcn_global_load_async_to_lds_b128(
          (gv4i_p)asrc, (lv4i_p)&s_anc[i][0], 0, 0);
    } else {
      s_box[i][0] = 0.f; s_box[i][1] = 0.f; s_box[i][2] = 0.f; s_box[i][3] = 0.f;
      s_area[i] = 0.f; s_sc[i] = 0.f;
    }
  }
  // overlap the 128KB mask zeroing with the in-flight async gathers
  for (int i = tid; i < KPAD * 32; i += 1024) mask[i] = 0;
#  ifdef HAVE_WAIT_ASYNC
  __builtin_amdgcn_s_wait_asynccnt(0);
#  else
  asm volatile("s_wait_asynccnt 0x0" ::: "memory");
#  endif
  // decode in place: deltas sit in s_box, anchors in s_anc (each thread owns its rows)
  for (int i = tid; i < KPRE; i += 1024) {
    float d0 = s_box[i][0], d1 = s_box[i][1], d2 = s_box[i][2], d3 = s_box[i][3];
    float a0 = s_anc[i][0], a1 = s_anc[i][1], a2 = s_anc[i][2], a3 = s_anc[i][3];
    float aw = a2 - a0, ah = a3 - a1;
    float acx = a0 + 0.5f * aw, acy = a1 + 0.5f * ah;
    float cx = d0 * aw + acx, cy = d1 * ah + acy;
    float w = expf(d2) * aw, h = expf(d3) * ah;
    float x1 = fminf(fmaxf(cx - 0.5f * w, 0.f), fw);
    float y1 = fminf(fmaxf(cy - 0.5f * h, 0.f), fh);
    float x2 = fminf(fmaxf(cx + 0.5f * w, 0.f), fw);
    float y2 = fminf(fmaxf(cy + 0.5f * h, 0.f), fh);
    s_box[i][0] = x1; s_box[i][1] = y1; s_box[i][2] = x2; s_box[i][3] = y2;
    s_area[i] = (x2 - x1) * (y2 - y1);
    s_sc[i] = sel_score[b * KPRE + i];
  }
#else
  for (int i = tid; i < KPAD; i += 1024) {
    float x1 = 0.f, y1 = 0.f, x2 = 0.f, y2 = 0.f, scv = 0.f;
    if (i < KPRE) {
      unsigned idx = sel_idx[b * KPRE + i];
      const float* d = deltas + ((size_t)b * NANCH + idx) * 4;
      const float* a = anchors + (size_t)idx * 4;
      float aw = a[2] - a[0], ah = a[3] - a[1];
      float acx = a[0] + 0.5f * aw, acy = a[1] + 0.5f * ah;
      float cx = d[0] * aw + acx, cy = d[1] * ah + acy;
      float w = expf(d[2]) * aw, h = expf(d[3]) * ah;
      x1 = fminf(fmaxf(cx - 0.5f * w, 0.f), fw);
      y1 = fminf(fmaxf(cy - 0.5f * h, 0.f), fh);
      x2 = fminf(fmaxf(cx + 0.5f * w, 0.f), fw);
      y2 = fminf(fmaxf(cy + 0.5f * h, 0.f), fh);
      scv = sel_score[b * KPRE + i];
    }
    s_box[i][0] = x1; s_box[i][1] = y1; s_box[i][2] = x2; s_box[i][3] = y2;
    s_area[i] = (x2 - x1) * (y2 - y1);
    s_sc[i] = scv;
  }
  for (int i = tid; i < KPAD * 32; i += 1024) mask[i] = 0;
#endif
  __syncthreads();

  const int wave = tid >> 5;
  const int lane = tid & 31;
  const int ln = lane & 15;
  const int hi = lane >> 4;

  // paired 16x16 tiles: (row block ti, word column wc) with wc >= ti/2.
  // One wave per pair -> EXEC all 1s around the WMMAs (pair selection is wave-uniform).
  int t = 0;
  for (int ti = 0; ti < NT; ++ti) {
    for (int wc = (ti >> 1); wc < 32; ++wc, ++t) {
      if ((t & 31) != wave) continue;
      const int colbase = wc * 32;
      const int col0 = colbase + ln;
      const int col1 = colbase + 16 + ln;
      const float4 cb0 = *reinterpret_cast<const float4*>(&s_box[col0][0]);
      const float4 cb1 = *reinterpret_cast<const float4*>(&s_box[col1][0]);
      float iv0[8], iv1[8];
      v8f C0, C1;
#pragma unroll
      for (int r = 0; r < 8; ++r) {
        int row = ti * 16 + r + hi * 8;
        const float4 rb = *reinterpret_cast<const float4*>(&s_box[row][0]);
        float w0 = fmaxf(fminf(rb.z, cb0.z) - fmaxf(rb.x, cb0.x), 0.f);
        float h0 = fmaxf(fminf(rb.w, cb0.w) - fmaxf(rb.y, cb0.y), 0.f);
        float w1 = fmaxf(fminf(rb.z, cb1.z) - fmaxf(rb.x, cb1.x), 0.f);
        float h1 = fmaxf(fminf(rb.w, cb1.w) - fmaxf(rb.y, cb1.y), 0.f);
        iv0[r] = w0 * h0;  C0[r] = -iv0[r];       // accumulate -inter in the WMMA
        iv1[r] = w1 * h1;  C1[r] = -iv1[r];
      }
      v2f A, B0, B1;
      A.x  = hi ? 0.f : s_area[ti * 16 + ln];     // K=0 row of A
      A.y  = hi ? 0.f : 1.0f;                     // K=1 row of A
      B0.x = hi ? 0.f : 1.0f;                     // K=0 row of B
      B0.y = hi ? 0.f : s_area[col0];             // K=1 row of B
      B1.x = hi ? 0.f : 1.0f;
      B1.y = hi ? 0.f : s_area[col1];
      // union tiles: D = area_i + area_j - inter
      v8f U0 = __builtin_amdgcn_wmma_f32_16x16x4_f32(false, A, false, B0,
                                                     (short)0, C0, false, false);
      v8f U1 = __builtin_amdgcn_wmma_f32_16x16x4_f32(false, A, false, B1,
                                                     (short)0, C1, false, false);
#pragma unroll
      for (int r = 0; r < 8; ++r) {
        int row = ti * 16 + r + hi * 8;
        bool p0 = (iv0[r] > NMSTHR * U0[r]) && (col0 > row) &&
                  (col0 < KPRE) && (row < KPRE);
        bool p1 = (iv1[r] > NMSTHR * U1[r]) && (col1 > row) &&
                  (col1 < KPRE) && (row < KPRE);
        unsigned long long bal0 = __ballot(p0);   // wave32: low 32 bits
        unsigned long long bal1 = __ballot(p1);
        unsigned w_lo = (unsigned)(bal0 & 0xFFFFull) |
                        ((unsigned)(bal1 & 0xFFFFull) << 16);          // row ti*16+r
        unsigned w_hi = (unsigned)((bal0 >> 16) & 0xFFFFull) |
                        ((unsigned)((bal1 >> 16) & 0xFFFFull) << 16);  // row +8
        if (lane == 0)       mask[(ti * 16 + r) * 32 + wc] = w_lo;
        else if (lane == 16) mask[(ti * 16 + r + 8) * 32 + wc] = w_hi;
      }
    }
  }
  __syncthreads();

  float* outb = out + (size_t)b * (KPOST * 5);
  for (int i = tid; i < KPOST * 5; i += 1024) outb[i] = 0.f;
  if (tid < 32) s_keep[tid] = (tid < 31) ? 0xFFFFFFFFu : 0x000000FFu;  // 1000 bits
  __syncthreads();

  // greedy scan: single wave, lane j owns keep word j
  if (tid < 32) {
    volatile unsigned* vk = s_keep;
    for (int i = 0; i < KPRE; ++i) {
      if ((vk[i >> 5] >> (i & 31)) & 1u) {        // uniform across the wave
        unsigned m = mask[i * 32 + lane];
        if (m) vk[lane] &= ~m;
      }
    }
  }
  __syncthreads();

  // kept boxes first in score order, rest already zeroed (matches reference padding)
  if (tid == 0) {
    int cnt = 0;
    for (int i = 0; i < KPRE && cnt < KPOST; ++i) {
      if ((s_keep[i >> 5] >> (i & 31)) & 1u) {
        float* o = outb + cnt * 5;
        o[0] = s_box[i][0]; o[1] = s_box[i][1];
        o[2] = s_box[i][2]; o[3] = s_box[i][3];
        o[4] = s_sc[i];
        ++cnt;
      }
    }
  }
}

extern "C" void kernel_launch(void* const* d_in, const int* in_sizes, int n_in,
                              void* d_out, int out_size, void* d_ws, size_t ws_size,
                              hipStream_t stream) {
  const float* cls  = (const float*)d_in[0];   // [32,90000,1]
  const float* reg  = (const float*)d_in[1];   // [32,90000,4]
  const float* anch = (const float*)d_in[2];   // [90000,4]
  const int* p_h    = (const int*)d_in[3];
  const int* p_w    = (const int*)d_in[4];
  float* out = (float*)d_out;                  // [32,300,5]

  // ws layout (~11.8 MB): [scores 32*90000 f32][sel_idx 32*1000 u32][sel_score 32*1000 f32]
  // mask (32*1024*32 u32 = 4 MB) ALIASES the scores region: scores are dead after
  // topk_select_kernel, and score_sigmoid_kernel fully rewrites them each call, so the
  // aliasing is replay-safe under graph capture.
  float* ws_scores    = (float*)d_ws;
  unsigned* ws_selidx = (unsigned*)(ws_scores + (size_t)BSZ * NANCH);
  float* ws_selsc     = (float*)(ws_selidx + BSZ * KPRE);
  unsigned* ws_mask   = (unsigned*)d_ws;

  int n4 = (BSZ * NANCH) / 4;
  score_sigmoid_kernel<<<(n4 + 255) / 256, 256, 0, stream>>>(
      (const float4*)cls, (float4*)ws_scores, n4);
  topk_select_kernel<<<BSZ, 1024, 0, stream>>>(ws_scores, ws_selidx, ws_selsc);
  nms_kernel<<<BSZ, 1024, 0, stream>>>(reg, anch, ws_selidx, ws_selsc,
                                       ws_mask, out, p_h, p_w);
}